// FourierDeformationPk_20469814133247
// MI455X (gfx1250) — compile-verified
//
#include <hip/hip_runtime.h>
#include <math.h>

// ---------------------------------------------------------------------------
// FourierDeformationPk for MI455X (gfx1250, wave32)
//   spectrum scatter -> hermitian fill -> 3D iRFFT (radix-16 via f32 WMMA)
//   -> trilinear border-clamped grid sample.
// ---------------------------------------------------------------------------

typedef __attribute__((ext_vector_type(2))) float v2f;
typedef __attribute__((ext_vector_type(8))) float v8f;

#define DZ 32
#define DY 256
#define DX 256
#define HX 129                       // rfft half-spectrum length along x
#define NSPAT (DZ * DY * DX)         // 2,097,152 voxels
#define PLANE (DZ * DY * HX)         // 1,056,768 (one re or im plane)
#define NFULL (2 * PLANE)            // per-channel re+im
#define PI2 6.28318530717958647692f

__device__ __forceinline__ int sidx(int c, int p, int z, int y, int x) {
  return c * NFULL + p * PLANE + (z * DY + y) * HX + x;
}
__device__ __forceinline__ int imin(int a, int b) { return a < b ? a : b; }

// ------------------------- spectrum construction ---------------------------

__global__ void k_zero(float* __restrict__ p, int n) {
  int i = blockIdx.x * blockDim.x + threadIdx.x;
  if (i < n) p[i] = 0.0f;
}

__global__ void k_scatter(const float* __restrict__ seeds,
                          const float* __restrict__ Pk,
                          const float* __restrict__ defsc,
                          const int* __restrict__ fidx,
                          float* __restrict__ spec, int fdim) {
  int tid = blockIdx.x * blockDim.x + threadIdx.x;
  if (tid >= 3 * fdim) return;
  int c = tid / fdim, j = tid - c * fdim;
  spec[c * NFULL + fidx[j]] = seeds[tid] * Pk[j] * defsc[c];
}

// Hermitian symmetrization of the x=0 and x=128 planes (m0=16, m1=128).
// Region A: z in 17..31, all y : f[z,y] = conj(f[32-z, (256-y)%256])
// Region B: z in {0,16}, y in 129..255 : f[z,y] = conj(f[z, 256-y])
__global__ void k_herm(float* __restrict__ spec) {
  const int per = 15 * 256 + 2 * 127;  // 4094 jobs per (c, xplane)
  int tid = blockIdx.x * blockDim.x + threadIdx.x;
  if (tid >= 3 * 2 * per) return;
  int j = tid % per, t = tid / per;
  int x = (t & 1) * 128, c = t >> 1;
  int zd, yd, zs, ys;
  if (j < 15 * 256) {
    zd = 17 + j / 256; yd = j % 256;
    zs = 32 - zd;      ys = (256 - yd) & 255;
  } else {
    int j2 = j - 15 * 256;
    zd = (j2 < 127) ? 0 : 16; yd = 129 + (j2 % 127);
    zs = zd;                  ys = 256 - yd;
  }
  float* re = spec + c * NFULL;
  float* im = re + PLANE;
  int di = (zd * DY + yd) * HX + x;
  int si = (zs * DY + ys) * HX + x;
  re[di] = re[si];
  im[di] = -im[si];
}

// ------------------------- z axis: 32-point iDFT ---------------------------
// One block per (c,z,y); LDS-cached twiddles; x lanes coalesced.
__global__ void k_fft_z(const float* __restrict__ in, float* __restrict__ out) {
  __shared__ float tc[32], ts[32];
  int bid = blockIdx.x;
  int y = bid % DY, t = bid / DY;
  int z = t % DZ, c = t / DZ;
  int x = threadIdx.x;
  if (threadIdx.x < 32) {
    float ang = (PI2 / 32.0f) * (float)((z * (int)threadIdx.x) & 31);
    float s, cc;
    __sincosf(ang, &s, &cc);
    ts[threadIdx.x] = s;
    tc[threadIdx.x] = cc;
  }
  __syncthreads();
  if (x < HX) {
    float ar = 0.0f, ai = 0.0f;
    for (int k = 0; k < DZ; ++k) {
      float cr = in[sidx(c, 0, k, y, x)];
      float ci = in[sidx(c, 1, k, y, x)];
      float wc = tc[k], ws = ts[k];
      ar = fmaf(cr, wc, fmaf(-ci, ws, ar));
      ai = fmaf(cr, ws, fmaf(ci, wc, ai));
    }
    out[sidx(c, 0, z, y, x)] = ar;
    out[sidx(c, 1, z, y, x)] = ai;
  }
}

// ------------------- 256-point iDFT via V_WMMA_F32_16X16X4 -----------------
// Complex MAC: Dr += Ar*Br - Ai*Bi ; Di += Ar*Bi + Ai*Br.
// f32 WMMA NEG bits only negate C, so negate Ai in VALU for the subtraction.
__device__ __forceinline__ void cmac16(v8f& accR, v8f& accI,
                                       v2f ar, v2f ai, v2f br, v2f bi) {
  v2f nai; nai.x = -ai.x; nai.y = -ai.y;
  accR = __builtin_amdgcn_wmma_f32_16x16x4_f32(false, ar, false, br, (short)0, accR, false, false);
  accR = __builtin_amdgcn_wmma_f32_16x16x4_f32(false, nai, false, bi, (short)0, accR, false, false);
  accI = __builtin_amdgcn_wmma_f32_16x16x4_f32(false, ar, false, bi, (short)0, accI, false, false);
  accI = __builtin_amdgcn_wmma_f32_16x16x4_f32(false, ai, false, br, (short)0, accI, false, false);
}

// A-fragment (16x4, 32-bit): lanes 0-15 -> M=lane, K = 4kb+{0,1};
//                            lanes 16-31 -> M=lane-16, K = 4kb+{2,3}.
__device__ __forceinline__ void dft16_afrag(int kb, int lane, v2f& ar, v2f& ai) {
  int m = lane & 15;
  int k0 = 4 * kb + 2 * (lane >> 4);
  float a0 = (PI2 / 16.0f) * (float)((m * k0) & 15);
  float a1 = (PI2 / 16.0f) * (float)((m * (k0 + 1)) & 15);
  float s0, c0, s1, c1;
  __sincosf(a0, &s0, &c0);
  __sincosf(a1, &s1, &c1);
  ar.x = c0; ai.x = s0;
  ar.y = c1; ai.y = s1;
}
// B-fragment (4x16): lane holds column N=lane&15; K mirrors the A half-split.
__device__ __forceinline__ void dft16_bfrag(int kb, int lane, v2f& br, v2f& bi) {
  int n = lane & 15;
  int k0 = 4 * kb + 2 * (lane >> 4);
  float a0 = (PI2 / 16.0f) * (float)((k0 * n) & 15);
  float a1 = (PI2 / 16.0f) * (float)(((k0 + 1) * n) & 15);
  float s0, c0, s1, c1;
  __sincosf(a0, &s0, &c0);
  __sincosf(a1, &s1, &c1);
  br.x = c0; bi.x = s0;
  br.y = c1; bi.y = s1;
}

// In-wave 256-point inverse DFT of LDS data re[n], im[n] (n = 16*n2 + n1):
//   S  = DFT16 x M            (M(n2,n1) = data[16*n2+n1])    -> WMMA stage 1
//   T  = S .* W256^{k1*n1}                                    (VALU twiddle)
//   D  = T x DFT16            (out[k1 + 16*k2] = D(k1,k2))    -> WMMA stage 2
// Intermediate T is exchanged across lanes through the same LDS buffers.
__device__ __forceinline__ void idft256_wave(float* re, float* im, int lane,
                                             v8f& outR, v8f& outI) {
  v8f s1R = {0.f,0.f,0.f,0.f,0.f,0.f,0.f,0.f};
  v8f s1I = s1R;
  // stage 1: A = DFT16, B = M (element (K=n2, N=n1) = data[16*K + N])
  for (int kb = 0; kb < 4; ++kb) {
    v2f ar, ai, br, bi;
    dft16_afrag(kb, lane, ar, ai);
    int N = lane & 15;
    int K0 = 4 * kb + 2 * (lane >> 4);
    br.x = re[16 * K0 + N];       br.y = re[16 * (K0 + 1) + N];
    bi.x = im[16 * K0 + N];       bi.y = im[16 * (K0 + 1) + N];
    cmac16(s1R, s1I, ar, ai, br, bi);
  }
  // twiddle (C/D layout: VGPR v, lane l -> M = v + 8*(l>=16), N = l%16)
  {
    int hi = lane >> 4, N = lane & 15;
    for (int v = 0; v < 8; ++v) {
      int k1 = v + 8 * hi;
      float s, c;
      __sincosf((PI2 / 256.0f) * (float)(k1 * N), &s, &c);
      float xr = s1R[v], xi = s1I[v];
      re[k1 * 16 + N] = xr * c - xi * s;
      im[k1 * 16 + N] = xr * s + xi * c;
    }
  }
  __syncthreads();
  // stage 2: A = T (M=k1, K=n1), B = DFT16 (K=n1, N=k2)
  outR = {0.f,0.f,0.f,0.f,0.f,0.f,0.f,0.f};
  outI = outR;
  for (int kb = 0; kb < 4; ++kb) {
    v2f ar, ai, br, bi;
    int m = lane & 15;
    int k0 = 4 * kb + 2 * (lane >> 4);
    ar.x = re[m * 16 + k0];       ar.y = re[m * 16 + k0 + 1];
    ai.x = im[m * 16 + k0];       ai.y = im[m * 16 + k0 + 1];
    dft16_bfrag(kb, lane, br, bi);
    cmac16(outR, outI, ar, ai, br, bi);
  }
}

// y axis: one wave per (c,z,x) column of 256 complex values.
__global__ void k_fft_y(const float* __restrict__ in, float* __restrict__ out) {
  __shared__ float lre[8][256];
  __shared__ float lim[8][256];
  int wv = blockIdx.x * 8 + (threadIdx.x >> 5);
  int lane = threadIdx.x & 31;
  int x = wv % HX, t = wv / HX;
  int z = t % DZ, c = t / DZ;
  float* re = lre[threadIdx.x >> 5];
  float* im = lim[threadIdx.x >> 5];
  for (int i = lane; i < 256; i += 32) {
    re[i] = in[sidx(c, 0, z, i, x)];
    im[i] = in[sidx(c, 1, z, i, x)];
  }
  __syncthreads();
  v8f oR, oI;
  idft256_wave(re, im, lane, oR, oI);
  int hi = lane >> 4, k2 = lane & 15;
  for (int v = 0; v < 8; ++v) {
    int yy = (v + 8 * hi) + 16 * k2;
    out[sidx(c, 0, z, yy, x)] = oR[v];
    out[sidx(c, 1, z, yy, x)] = oI[v];
  }
}

// x axis: one wave per (c,z,y) row; expand 129 -> 256 by conjugate symmetry,
// 256-point iDFT, keep the real part, apply 1/(32*256*256).
__global__ void k_irfft_x(const float* __restrict__ in, float* __restrict__ out) {
  __shared__ float lre[8][256];
  __shared__ float lim[8][256];
  int wv = blockIdx.x * 8 + (threadIdx.x >> 5);
  int lane = threadIdx.x & 31;
  int y = wv % DY, t = wv / DY;
  int z = t % DZ, c = t / DZ;
  float* re = lre[threadIdx.x >> 5];
  float* im = lim[threadIdx.x >> 5];
  const float* pre = in + sidx(c, 0, z, y, 0);
  const float* pim = in + sidx(c, 1, z, y, 0);
  for (int i = lane; i < HX; i += 32)       { re[i] = pre[i];        im[i] =  pim[i]; }
  for (int i = HX + lane; i < 256; i += 32) { re[i] = pre[256 - i];  im[i] = -pim[256 - i]; }
  __syncthreads();
  v8f oR, oI;
  idft256_wave(re, im, lane, oR, oI);
  (void)oI;
  const float norm = 1.0f / (float)((long)DZ * DY * DX);
  float* po = out + ((c * DZ + z) * DY + y) * DX;
  int hi = lane >> 4, k2 = lane & 15;
  for (int v = 0; v < 8; ++v) po[(v + 8 * hi) + 16 * k2] = oR[v] * norm;
}

// --------------------- trilinear border-clamped sampler --------------------
// NOTE: disp may live inside d_out (disjoint only per-voxel); no restrict on
// disp/out so the compiler keeps disp loads ahead of the output stores.
__global__ void k_interp(const float* __restrict__ fr,
                         const float* __restrict__ grid,
                         const float* disp, float* out) {
  int x = threadIdx.x;
  int zy = blockIdx.x;
  int y = zy % DY, z = zy / DY;
  int si = (z * DY + y) * DX + x;
  float gz = grid[0 * NSPAT + si], gy = grid[1 * NSPAT + si], gx = grid[2 * NSPAT + si];
  float dz = disp[0 * NSPAT + si], dy = disp[1 * NSPAT + si], dxv = disp[2 * NSPAT + si];
  const float nz = (float)(DZ - 1), ny = (float)(DY - 1), nx = (float)(DX - 1);
  // moved = 2*((g - d)/norm) - 1 ; sample coord = clip((moved+1)*0.5*(N-1))
  float fz = fminf(fmaxf(((2.0f * ((gz - dz) / nz) - 1.0f) + 1.0f) * 0.5f * nz, 0.0f), nz);
  float fy = fminf(fmaxf(((2.0f * ((gy - dy) / ny) - 1.0f) + 1.0f) * 0.5f * ny, 0.0f), ny);
  float fx = fminf(fmaxf(((2.0f * ((gx - dxv) / nx) - 1.0f) + 1.0f) * 0.5f * nx, 0.0f), nx);
  float z0f = floorf(fz), y0f = floorf(fy), x0f = floorf(fx);
  float wz = fz - z0f, wy = fy - y0f, wx = fx - x0f;
  int z0 = (int)z0f, y0 = (int)y0f, x0 = (int)x0f;
  int z1 = imin(z0 + 1, DZ - 1), y1 = imin(y0 + 1, DY - 1), x1 = imin(x0 + 1, DX - 1);
  int i000 = (z0 * DY + y0) * DX + x0, i001 = (z0 * DY + y0) * DX + x1;
  int i010 = (z0 * DY + y1) * DX + x0, i011 = (z0 * DY + y1) * DX + x1;
  int i100 = (z1 * DY + y0) * DX + x0, i101 = (z1 * DY + y0) * DX + x1;
  int i110 = (z1 * DY + y1) * DX + x0, i111 = (z1 * DY + y1) * DX + x1;
  for (int bc = 0; bc < 12; ++bc) {
    const float* f = fr + (size_t)bc * NSPAT;
    float c00 = f[i000] * (1.0f - wx) + f[i001] * wx;
    float c01 = f[i010] * (1.0f - wx) + f[i011] * wx;
    float c10 = f[i100] * (1.0f - wx) + f[i101] * wx;
    float c11 = f[i110] * (1.0f - wx) + f[i111] * wx;
    float c0 = c00 * (1.0f - wy) + c01 * wy;
    float c1 = c10 * (1.0f - wy) + c11 * wy;
    out[(size_t)bc * NSPAT + si] = c0 * (1.0f - wz) + c1 * wz;
  }
}

// --------------------------------- launch ----------------------------------
extern "C" void kernel_launch(void* const* d_in, const int* in_sizes, int n_in,
                              void* d_out, int out_size, void* d_ws, size_t ws_size,
                              hipStream_t stream) {
  const float* fr    = (const float*)d_in[0];
  const float* seeds = (const float*)d_in[1];
  const float* Pk    = (const float*)d_in[2];
  const float* defsc = (const float*)d_in[3];
  const float* grid  = (const float*)d_in[4];
  const int*   fidx  = (const int*)d_in[5];
  int fdim = in_sizes[2];

  // Workspace plan (keeps d_ws requirement to one spectrum buffer, ~25.4 MB):
  //   spec0 : d_ws                           (3*NFULL floats)
  //   spec1 : d_out[0 ..)                    (3*NFULL floats, scratch)
  //   disp  : d_out[9*NSPAT ..)              (3*NSPAT floats, scratch)
  // k_interp fully overwrites d_out last; per-voxel disp reads precede the
  // aliased stores within each thread (no restrict on those pointers).
  float* spec0 = (float*)d_ws;
  float* outp  = (float*)d_out;
  float* spec1 = outp;
  float* disp  = outp + (size_t)9 * NSPAT;

  const int nz = 3 * NFULL;
  k_zero<<<dim3((nz + 255) / 256), dim3(256), 0, stream>>>(spec0, nz);
  k_scatter<<<dim3((3 * fdim + 255) / 256), dim3(256), 0, stream>>>(
      seeds, Pk, defsc, fidx, spec0, fdim);
  k_herm<<<dim3((3 * 2 * 4094 + 255) / 256), dim3(256), 0, stream>>>(spec0);
  k_fft_z<<<dim3(3 * DZ * DY), dim3(160), 0, stream>>>(spec0, spec1);
  k_fft_y<<<dim3(3 * DZ * HX / 8), dim3(256), 0, stream>>>(spec1, spec0);
  k_irfft_x<<<dim3(3 * DZ * DY / 8), dim3(256), 0, stream>>>(spec0, disp);
  k_interp<<<dim3(DZ * DY), dim3(256), 0, stream>>>(fr, grid, disp, outp);
}